// First_n_Last_51041391345711
// MI455X (gfx1250) — compile-verified
//
#include <hip/hip_runtime.h>

// ---------------------------------------------------------------------------
// Fused span-extraction (windowed max at span start/end) + f32 WMMA GEMM
// out[b,n,h] = sum_k span_reps[b,n,k] * W[h,k] + bias[h]
//
// Pass 1: tiled transpose W (768,1536) -> Wt (1536,768) in workspace, so the
//         WMMA B-fragment loads are fully coalesced (16 lanes -> 16
//         consecutive floats) instead of 16 rows 6KB apart.
// Pass 2: per 16-span row tile, gather the windowed-max span reps into LDS
//         (K chunks of 128), then V_WMMA_F32_16X16X4_F32 with one A-frag
//         reused across 6 B-frags per wave (8 waves x 96 columns = 768).
// ---------------------------------------------------------------------------

typedef float v2f __attribute__((ext_vector_type(2)));
typedef float v8f __attribute__((ext_vector_type(8)));

#define BATCH        4
#define SEQ          512
#define HID          768
#define K2           1536               // 2*HID (GEMM K)
#define NSPANS       4096
#define TOTAL_SPANS  (BATCH * NSPANS)   // 16384 (GEMM M)
#define KC           128                // K chunk staged in LDS
#define LDA          (KC + 4)           // padded LDS row stride (bank spread)
#define WT_BYTES     ((size_t)K2 * HID * sizeof(float))

// ---------------- Pass 1: W (HID x K2) -> Wt (K2 x HID) --------------------
__global__ __launch_bounds__(256) void transpose_w_kernel(
    const float* __restrict__ W, float* __restrict__ Wt)
{
    __shared__ float tile[32][33];                 // +1 pad: conflict-free
    const int k0 = blockIdx.x * 32;                // along K2
    const int h0 = blockIdx.y * 32;                // along HID
    const int tx = threadIdx.x & 31;
    const int ty = threadIdx.x >> 5;               // 8 rows/pass
    #pragma unroll
    for (int i = 0; i < 32; i += 8)                // coalesced read along k
        tile[ty + i][tx] = W[(size_t)(h0 + ty + i) * K2 + (k0 + tx)];
    __syncthreads();
    #pragma unroll
    for (int i = 0; i < 32; i += 8)                // coalesced write along h
        Wt[(size_t)(k0 + ty + i) * HID + (h0 + tx)] = tile[tx][ty + i];
}

// ---------------- Pass 2: fused gather + WMMA GEMM -------------------------
// USE_WT=1: B frags from transposed Wt (coalesced). USE_WT=0: direct from W.
template <bool USE_WT>
__global__ __launch_bounds__(256) void span_gemm_kernel(
    const float* __restrict__ token_reps,
    const int*   __restrict__ span_ids,
    const int*   __restrict__ span_masks,
    const int*   __restrict__ pooling,
    const float* __restrict__ W,       // (HID, K2) row-major
    const float* __restrict__ Wt,      // (K2, HID) row-major (if USE_WT)
    const float* __restrict__ bias,
    float*       __restrict__ out)
{
    __shared__ float aTile[16 * LDA];        // 16 spans x KC k-slice (padded)
    __shared__ int   sStart[16];
    __shared__ int   sEnd[16];
    __shared__ int   sMask[16];

    const int tid      = threadIdx.x;
    const int rowTile  = blockIdx.x;         // 0..1023
    const int spanBase = rowTile * 16;

    if (tid < 16) {
        const int s  = spanBase + tid;
        sStart[tid]  = span_ids[2 * s + 0];
        sEnd[tid]    = span_ids[2 * s + 1];
        sMask[tid]   = span_masks[s];
    }
    const int win = (pooling[0] != 0) ? 1 : 3;

    const int lane = tid & 31;               // wave32
    const int wave = tid >> 5;               // 8 waves
    const int cl   = lane & 15;              // M (A) / N (B,C) index
    const int kh   = lane >> 4;              // K-half select

    v8f acc[6];
    #pragma unroll
    for (int t = 0; t < 6; ++t) {
        #pragma unroll
        for (int i = 0; i < 8; ++i) acc[t][i] = 0.0f;
    }

    for (int k0 = 0; k0 < K2; k0 += KC) {
        __syncthreads();                     // LDS safe to overwrite

        // ---- cooperative gather of the 16 x KC span-rep tile into LDS ----
        #pragma unroll
        for (int i = 0; i < (16 * KC) / 256; ++i) {   // 8 elements/thread
            const int e  = tid + i * 256;
            const int m  = e >> 7;            // e / KC
            const int k  = e & (KC - 1);
            const int kg = k0 + k;
            float v;
            if (sMask[m]) {
                const int st = sStart[m];
                const int en = sEnd[m];
                const int s  = spanBase + m;
                const int bb = s >> 12;       // s / NSPANS
                const float* tp = token_reps + (size_t)bb * (SEQ * HID);
                v = -__builtin_inff();
                if (kg < HID) {               // start-window max, column kg
                    const int c = kg;
                    for (int j = 0; j < win; ++j) {
                        const int tk = st + j;
                        if (tk < en) v = fmaxf(v, tp[tk * HID + c]);
                    }
                } else {                      // end-window max, column kg-HID
                    const int c = kg - HID;
                    for (int j = 0; j < win; ++j) {
                        const int tk = en - win + j;
                        if (tk >= st) v = fmaxf(v, tp[tk * HID + c]);
                    }
                }
            } else {
                v = 0.0f;                     // masked span -> zero reps
            }
            aTile[m * LDA + k] = v;
        }
        __syncthreads();

        // ---- WMMA over this K chunk: 1 A-frag reused by 6 B-frags --------
        for (int kk = 0; kk < KC; kk += 4) {
            // A 16x4 f32 frag: lanes 0-15 rows, VGPR0/1 = K {0,1} (+2 upper)
            const v2f a = *(const v2f*)&aTile[cl * LDA + kk + 2 * kh];
            const int kr = k0 + kk + 2 * kh;  // first K row of this lane
            #pragma unroll
            for (int t = 0; t < 6; ++t) {
                const int col = wave * 96 + t * 16 + cl;
                v2f bf;
                if (USE_WT) {
                    // B[k][n] = Wt[k][col]: lanes 0-15 read 16 consecutive
                    // floats per row -> fully coalesced b32 loads
                    const float* wt0 = Wt + (size_t)kr * HID + col;
                    bf[0] = wt0[0];
                    bf[1] = wt0[HID];
                } else {
                    // fallback: strided b64 from W rows
                    bf = *(const v2f*)(W + (size_t)col * K2 + kr);
                }
                acc[t] = __builtin_amdgcn_wmma_f32_16x16x4_f32(
                    false, a, false, bf, (short)0, acc[t], false, false);
            }
        }
    }

    // ---- epilogue: bias add + store (C/D layout: lane=N, vgpr r = M row) --
    #pragma unroll
    for (int t = 0; t < 6; ++t) {
        const int col  = wave * 96 + t * 16 + cl;
        const float bv = bias[col];
        const int row0 = spanBase + kh * 8;
        #pragma unroll
        for (int r = 0; r < 8; ++r) {
            out[(size_t)(row0 + r) * HID + col] = acc[t][r] + bv;
        }
    }
}

extern "C" void kernel_launch(void* const* d_in, const int* in_sizes, int n_in,
                              void* d_out, int out_size, void* d_ws, size_t ws_size,
                              hipStream_t stream) {
    (void)in_sizes; (void)n_in; (void)out_size;
    const float* token_reps = (const float*)d_in[0];
    const int*   span_ids   = (const int*)d_in[1];
    const int*   span_masks = (const int*)d_in[2];
    const int*   pooling    = (const int*)d_in[3];
    const float* W          = (const float*)d_in[4];
    const float* b          = (const float*)d_in[5];
    float*       out        = (float*)d_out;

    dim3 grid(TOTAL_SPANS / 16);   // 1024 row tiles
    dim3 block(256);               // 8 wave32 waves

    if (d_ws != nullptr && ws_size >= WT_BYTES) {
        float* Wt = (float*)d_ws;
        transpose_w_kernel<<<dim3(K2 / 32, HID / 32), 256, 0, stream>>>(W, Wt);
        span_gemm_kernel<true><<<grid, block, 0, stream>>>(
            token_reps, span_ids, span_masks, pooling, W, Wt, b, out);
    } else {
        span_gemm_kernel<false><<<grid, block, 0, stream>>>(
            token_reps, span_ids, span_masks, pooling, W, nullptr, b, out);
    }
}